// GraphConvLayer_39582418600193
// MI455X (gfx1250) — compile-verified
//
#include <hip/hip_runtime.h>

typedef __attribute__((ext_vector_type(2))) float v2f;
typedef __attribute__((ext_vector_type(8))) float v8f;

#define DIM 64

// ---------------- k0: deg = 1.0 (self loop contribution) ----------------
__global__ void k_deg_init(float* __restrict__ deg, int N) {
    int i = blockIdx.x * blockDim.x + threadIdx.x;
    if (i < N) deg[i] = 1.0f;
}

// ---------------- k1: deg[col[e]] += 1 ----------------
__global__ void k_deg_accum(const int* __restrict__ col, float* __restrict__ deg, int E) {
    int e = blockIdx.x * blockDim.x + threadIdx.x;
    if (e < E) atomicAdd(&deg[col[e]], 1.0f);
}

// ---------------- k2: deg -> rsqrt(deg) in place ----------------
__global__ void k_dinv(float* __restrict__ deg, int N) {
    int i = blockIdx.x * blockDim.x + threadIdx.x;
    if (i < N) deg[i] = rsqrtf(deg[i]);
}

// ---------------- k3: xw = x @ W^T via V_WMMA_F32_16X16X4_F32 ----------------
// One wave computes a 16(nodes) x 16(out dims) tile; 16 WMMAs chain over K=64.
// A-frag (16x4 f32): lane holds x[nb + (lane&15)][k + (lane>>4)*2 .. +1]
// B-frag (4x16 f32): B[k][n] = W[db+n][k] -> lane holds W[db + (lane&15)][k + (lane>>4)*2 .. +1]
// D (16x16 f32, 8 VGPRs): vgpr v -> row M = v + (lane>>4)*8, col N = lane&15
__global__ void k_gemm_wmma(const float* __restrict__ x,
                            const float* __restrict__ W,
                            float* __restrict__ xw, int N) {
    const int lane  = threadIdx.x & 31;
    const int wave  = threadIdx.x >> 5;        // 0..3 -> output-dim tile
    const int nb    = blockIdx.x * 16;          // node tile base (N % 16 == 0)
    const int db    = wave * 16;                // out-dim tile base
    const int r     = lane & 15;
    const int khalf = (lane >> 4) << 1;         // 0 or 2

    const float* xrow = x + (size_t)(nb + r) * DIM;
    const float* wrow = W + (size_t)(db + r) * DIM;

    v8f acc = {};
#pragma unroll
    for (int k = 0; k < DIM; k += 4) {
        const int kk = k + khalf;
        v2f a = *(const v2f*)(xrow + kk);
        v2f b = *(const v2f*)(wrow + kk);
        // 8 args: (neg_a, A, neg_b, B, c_mod, C, reuse_a, reuse_b)
        acc = __builtin_amdgcn_wmma_f32_16x16x4_f32(
            false, a, false, b, (short)0, acc, false, false);
    }

    const int mhi = (lane >> 4) << 3;           // 0 or 8
#pragma unroll
    for (int v = 0; v < 8; ++v) {
        xw[(size_t)(nb + v + mhi) * DIM + db + r] = acc[v];
    }
}

// ---------------- k4: out[n][d] = b[d] + dinv[n]^2 * xw[n][d] ----------------
__global__ void k_init_out(const float* __restrict__ xw, const float* __restrict__ dinv,
                           const float* __restrict__ bias, float* __restrict__ out, int N) {
    int t = blockIdx.x * blockDim.x + threadIdx.x;   // over N*32 (float2 per thread)
    if (t >= N * 32) return;
    int n  = t >> 5;
    int d2 = (t & 31) << 1;
    float di = dinv[n];
    float s  = di * di;
    float2 m = *(const float2*)(xw + (size_t)n * DIM + d2);
    float2 o;
    o.x = bias[d2]     + s * m.x;
    o.y = bias[d2 + 1] + s * m.y;
    *(float2*)(out + (size_t)n * DIM + d2) = o;
}

// ---------------- k5: edge scatter: out[col] += dinv[row]*dinv[col] * xw[row] ----------------
// One wave32 per edge; each lane handles 2 dims (float2 gather, 2x f32 L2 atomics).
__global__ void k_scatter(const int* __restrict__ row, const int* __restrict__ col,
                          const float* __restrict__ xw, const float* __restrict__ dinv,
                          float* __restrict__ out, int E) {
    int t = blockIdx.x * blockDim.x + threadIdx.x;   // over E*32
    int e = t >> 5;
    if (e >= E) return;
    int lane = t & 31;
    int r = row[e];
    int c = col[e];
    float norm = dinv[r] * dinv[c];
    float2 m = *(const float2*)(xw + (size_t)r * DIM + (lane << 1));
    float* dst = out + (size_t)c * DIM + (lane << 1);
    atomicAdd(dst,     norm * m.x);
    atomicAdd(dst + 1, norm * m.y);
}

// ---------------- k6: ReLU in place ----------------
__global__ void k_relu(float* __restrict__ out, int total4) {
    int i = blockIdx.x * blockDim.x + threadIdx.x;   // over total/4 (float4)
    if (i >= total4) return;
    float4 v = *(float4*)(out + (size_t)i * 4);
    v.x = fmaxf(v.x, 0.0f);
    v.y = fmaxf(v.y, 0.0f);
    v.z = fmaxf(v.z, 0.0f);
    v.w = fmaxf(v.w, 0.0f);
    *(float4*)(out + (size_t)i * 4) = v;
}

extern "C" void kernel_launch(void* const* d_in, const int* in_sizes, int n_in,
                              void* d_out, int out_size, void* d_ws, size_t ws_size,
                              hipStream_t stream) {
    const float* x  = (const float*)d_in[0];
    const int*   ei = (const int*)d_in[1];    // [2, E] flat: row then col
    const float* W  = (const float*)d_in[2];
    const float* b  = (const float*)d_in[3];
    float* out = (float*)d_out;

    const int N = in_sizes[0] / DIM;          // 100000
    const int E = in_sizes[1] / 2;            // 1600000
    const int* row = ei;
    const int* col = ei + E;

    // Workspace layout: [dinv: N floats][pad][xw: N*DIM floats]  (~26 MB)
    float* dinv = (float*)d_ws;
    float* xw   = dinv + ((N + 3) & ~3);      // 16B-aligned

    // 1) degree (self-loop = 1.0 baseline)
    k_deg_init<<<(N + 255) / 256, 256, 0, stream>>>(dinv, N);
    k_deg_accum<<<(E + 255) / 256, 256, 0, stream>>>(col, dinv, E);
    // 2) dinv = rsqrt(deg)
    k_dinv<<<(N + 255) / 256, 256, 0, stream>>>(dinv, N);
    // 3) xw = x @ W^T  (WMMA; N is an exact multiple of 16)
    k_gemm_wmma<<<N / 16, 128, 0, stream>>>(x, W, xw, N);
    // 4) out = b + dinv^2 * xw   (self-loop message + bias)
    k_init_out<<<(N * 32 + 255) / 256, 256, 0, stream>>>(xw, dinv, b, out, N);
    // 5) scatter-add edge messages (wave per edge, L2-resident atomics)
    {
        long long threads = (long long)E * 32;
        int grid = (int)((threads + 255) / 256);
        k_scatter<<<grid, 256, 0, stream>>>(row, col, xw, dinv, out, E);
    }
    // 6) ReLU
    {
        int total4 = (N * DIM) / 4;
        k_relu<<<(total4 + 255) / 256, 256, 0, stream>>>(out, total4);
    }
}